// CustomPositionsPiecewiseConv2d_22351009808401
// MI455X (gfx1250) — compile-verified
//
#include <hip/hip_runtime.h>

typedef __attribute__((ext_vector_type(2))) float v2f;
typedef __attribute__((ext_vector_type(8))) float v8f;

namespace {
constexpr int C_IN  = 32;
constexpr int H     = 64;
constexpr int W     = 64;
constexpr int O_OUT = 128;
constexpr int P     = 5;
constexpr int K2    = 9;    // 3x3 kernel positions
constexpr int BM    = 128;  // pixel rows per workgroup tile
constexpr int KC    = 56;   // per-channel K chunk: 45 coeff + 9 ident + 2 pad
constexpr int ASTR  = 58;   // LDS row stride for A (even, conflict-free mod 64)
constexpr int BSTR2 = 288;  // floats per B row-pair: 128 n * 2 interleaved + pad (288 % 64 == 32)
constexpr float ATOLC = 1e-5f;
constexpr float RTOLC = 1e-5f;
}

__global__ __launch_bounds__(256) void ppconv_wmma(
    const float* __restrict__ x, const float* __restrict__ wgt,
    const float* __restrict__ bias, const float* __restrict__ pos,
    float* __restrict__ out)
{
  __shared__ float As[BM * ASTR];            // [pixel-row][kk]            coeff tile
  __shared__ float Bs[(KC / 2) * BSTR2];     // [kk/2][n][2] interleaved   weight tile

  const int t    = threadIdx.x;
  const int wave = t >> 5;
  const int lane = t & 31;
  const int half = lane >> 4;
  const int ln   = lane & 15;
  const int wr   = wave >> 2;   // wave row   (0..1)  -> 64 pixel rows
  const int wc   = wave & 3;    // wave col   (0..3)  -> 32 outputs

  // positions + per-segment affine transforms (uniform across threads)
  float pv[P];
#pragma unroll
  for (int p = 0; p < P; ++p) pv[p] = pos[p];
  float varc[P - 1], cst[P - 1];
#pragma unroll
  for (int p = 0; p < P - 1; ++p) {
    float d = pv[p + 1] - pv[p];
    d = (d == 0.0f) ? 1e-6f : d;
    varc[p] = 1.0f / d;
    cst[p]  = -pv[p] * varc[p];
  }

  // zero the constant pad lanes once: A cols 54,55 ; B row-pair 27 (rows 54,55)
  for (int idx = t; idx < BM * 2; idx += 256)
    As[(idx >> 1) * ASTR + 54 + (idx & 1)] = 0.0f;
  for (int idx = t; idx < 2 * O_OUT; idx += 256)
    Bs[27 * BSTR2 + (idx & 127) * 2 + (idx >> 7)] = 0.0f;

  const int gbase = blockIdx.x * BM;

  v8f acc[8];   // [im*2 + in] : im in 0..3 (16-row tiles), in in 0..1 (16-col tiles)
#pragma unroll
  for (int a = 0; a < 8; ++a)
#pragma unroll
    for (int e = 0; e < 8; ++e) acc[a][e] = 0.0f;

  for (int c = 0; c < C_IN; ++c) {
    __syncthreads();  // prior WMMA reads complete before LDS overwrite

    // warm next channel's input rows into cache (global_prefetch_b8)
    if (c + 1 < C_IN) {
      const int gg = gbase + (t & 127);
      const int pb = gg >> 12, prm = gg & 4095;
      __builtin_prefetch(&x[(((size_t)(pb * C_IN + (c + 1)) * H) << 6) + prm], 0, 1);
    }

    // ---- A tile: piecewise-linear coefficients + raw-v columns ----------
    for (int idx = t; idx < BM * K2; idx += 256) {
      const int r  = idx / K2;
      const int k  = idx - r * K2;
      const int g  = gbase + r;
      const int b  = g >> 12;
      const int rm = g & 4095;
      const int oh = rm >> 6, ow = rm & 63;
      const int i  = k / 3, j = k % 3;
      const int ih = oh + i - 1, iw = ow + j - 1;
      float v = 0.0f;
      if (ih >= 0 && ih < H && iw >= 0 && iw < W)
        v = x[(((b * C_IN + c) * H + ih) * W) + iw];

      float co[P] = {0.0f, 0.0f, 0.0f, 0.0f, 0.0f};
      const bool mf  = fabsf(v - pv[0])     <= ATOLC + RTOLC * fabsf(pv[0]);
      const bool ml  = fabsf(v - pv[P - 1]) <= ATOLC + RTOLC * fabsf(pv[P - 1]);
      const bool inr = !(mf || ml) && (v >= pv[0]) && (v <= pv[P - 1]);
      co[0]     = mf ? 1.0f : 0.0f;
      co[P - 1] = ml ? 1.0f : 0.0f;
#pragma unroll
      for (int p = 0; p < P - 1; ++p) {
        const float m  = (inr && v >= pv[p] && v < pv[p + 1]) ? 1.0f : 0.0f;
        const float tt = v * varc[p] + cst[p];
        co[p]     += m * (1.0f - tt);
        co[p + 1] += m * tt;
      }
      float* arow = &As[r * ASTR];
#pragma unroll
      for (int p = 0; p < P; ++p) arow[k * P + p] = co[p];
      arow[45 + k] = v;  // raw value column for the ident correction
    }

    // ---- B tile: Wk_eff (zeroed when ident) + ident indicator rows ------
    // Interleaved pair layout: Bs[(row>>1)*BSTR2 + n*2 + (row&1)]
    for (int idx = t; idx < O_OUT * K2; idx += 256) {
      const int o = idx & 127;
      const int k = idx >> 7;
      const int i = k / 3, j = k % 3;
      float wv[P];
      bool ident = true;
#pragma unroll
      for (int p = 0; p < P; ++p) {
        wv[p] = wgt[((((o * C_IN + c) * P + p) * 3 + i) * 3) + j];
        ident = ident && (fabsf(wv[p] - 1.0f) <= (ATOLC + RTOLC));
      }
#pragma unroll
      for (int p = 0; p < P; ++p) {
        const int row = k * P + p;
        Bs[(row >> 1) * BSTR2 + o * 2 + (row & 1)] = ident ? 0.0f : wv[p];
      }
      const int rr = 45 + k;
      Bs[(rr >> 1) * BSTR2 + o * 2 + (rr & 1)] = ident ? 1.0f : 0.0f;
    }

    __syncthreads();

    // ---- fp32 WMMA over the 56-deep K chunk -----------------------------
    const float* arow0 = &As[(wr * 64 + ln) * ASTR];
#pragma unroll
    for (int kk = 0; kk < KC; kk += 4) {
      const int kb = kk + 2 * half;       // K pair handled by this half-wave
      const int kp = (kk >> 1) + half;    // B row-pair index (kb/2)

      v2f afr[4];
#pragma unroll
      for (int im = 0; im < 4; ++im)
        afr[im] = *(const v2f*)(arow0 + im * 16 * ASTR + kb);   // A[m][kb..kb+1]

      const float* brow = &Bs[kp * BSTR2];
      v2f bfr[2];
#pragma unroll
      for (int in = 0; in < 2; ++in)
        bfr[in] = *(const v2f*)(brow + (wc * 32 + in * 16 + ln) * 2); // B[kb..kb+1][n]

#pragma unroll
      for (int im = 0; im < 4; ++im)
#pragma unroll
        for (int in = 0; in < 2; ++in)
          acc[im * 2 + in] = __builtin_amdgcn_wmma_f32_16x16x4_f32(
              false, afr[im], false, bfr[in], (short)0, acc[im * 2 + in],
              false, false);
    }
  }

  // ---- epilogue: bias add + coalesced float4 stores ---------------------
  // acc[im*2+in][r] is pixel (gbase + wr*64 + im*16 + 8*half + r),
  // output o = wc*32 + in*16 + ln.
#pragma unroll
  for (int im = 0; im < 4; ++im) {
    const int g0 = gbase + wr * 64 + im * 16 + 8 * half;  // multiple of 8
    const int b  = g0 >> 12;
    const int rm = g0 & 4095;
#pragma unroll
    for (int in = 0; in < 2; ++in) {
      const int o    = wc * 32 + in * 16 + ln;
      const float bv = bias[o];
      const v8f av   = acc[im * 2 + in];
      float* op = out + (((size_t)(b * O_OUT + o)) << 12) + rm;
      float4 v0 = make_float4(av[0] + bv, av[1] + bv, av[2] + bv, av[3] + bv);
      float4 v1 = make_float4(av[4] + bv, av[5] + bv, av[6] + bv, av[7] + bv);
      *(float4*)(op)     = v0;
      *(float4*)(op + 4) = v1;
    }
  }
}

extern "C" void kernel_launch(void* const* d_in, const int* in_sizes, int n_in,
                              void* d_out, int out_size, void* d_ws, size_t ws_size,
                              hipStream_t stream) {
  (void)in_sizes; (void)n_in; (void)d_ws; (void)ws_size; (void)out_size;
  const float* x    = (const float*)d_in[0];
  const float* wgt  = (const float*)d_in[1];
  const float* bias = (const float*)d_in[2];
  const float* pos  = (const float*)d_in[3];
  float* out        = (float*)d_out;

  const int M = 16 * 64 * 64;          // 65536 output pixels
  dim3 grid(M / BM);                   // 512 workgroups
  dim3 block(256);                     // 8 waves (wave32)
  hipLaunchKernelGGL(ppconv_wmma, grid, block, 0, stream, x, wgt, bias, pos, out);
}